// GAT_59390807769649
// MI455X (gfx1250) — compile-verified
//
#include <hip/hip_runtime.h>
#include <math.h>

#define NNODES 100000
#define NEDGES 1600000
#define IN_F   128
#define HIDF   16
#define HEADS  4
#define OUTF   32
#define NEG_SLOPE 0.2f
#define ORD_NEG_INF 0x007FFFFFu

typedef __attribute__((ext_vector_type(2))) float v2f;
typedef __attribute__((ext_vector_type(8))) float v8f;

__device__ __forceinline__ float leaky(float v) { return v > 0.f ? v : NEG_SLOPE * v; }
// order-preserving float<->uint map for atomic float max
__device__ __forceinline__ unsigned f2o(float f) {
    unsigned b = __float_as_uint(f);
    return (b & 0x80000000u) ? ~b : (b | 0x80000000u);
}
__device__ __forceinline__ float o2f(unsigned b) {
    return __uint_as_float((b & 0x80000000u) ? (b & 0x7FFFFFFFu) : ~b);
}

// ---------------- init ----------------
__global__ void k_init1(float* agg1, unsigned* m1, float* den1, float* accum) {
    int i = blockIdx.x * blockDim.x + threadIdx.x;
    int stride = gridDim.x * blockDim.x;
    for (int t = i; t < NNODES * 64; t += stride) agg1[t] = 0.f;
    for (int t = i; t < NNODES * HEADS; t += stride) { m1[t] = ORD_NEG_INF; den1[t] = 0.f; }
    if (i < 32) accum[i] = 0.f;
}
__global__ void k_init2(float* agg2, unsigned* m2, float* den2) {
    int i = blockIdx.x * blockDim.x + threadIdx.x;
    int stride = gridDim.x * blockDim.x;
    for (int t = i; t < NNODES * OUTF; t += stride) agg2[t] = 0.f;
    for (int t = i; t < NNODES; t += stride) { m2[t] = ORD_NEG_INF; den2[t] = 0.f; }
}

// ---------------- GEMM1: h1[N,64] = x[N,128] @ W1[128,64]  (fp32 WMMA) ----------------
__global__ __launch_bounds__(256) void k_gemm1(const float* __restrict__ x,
                                               const float* __restrict__ W1,
                                               float* __restrict__ h1) {
    __shared__ float xs[32][IN_F + 1];   // +1 pad: conflict-free column reads
    __shared__ float wt[IN_F][64 + 1];
    const int rowBase = blockIdx.x * 32;
    const int tid = threadIdx.x;
    for (int idx = tid; idx < 32 * IN_F; idx += 256) {
        int r = idx >> 7, c = idx & 127;
        int gr = rowBase + r;
        xs[r][c] = (gr < NNODES) ? x[gr * IN_F + c] : 0.f;
    }
    for (int idx = tid; idx < IN_F * 64; idx += 256) {
        int r = idx >> 6, c = idx & 63;
        wt[r][c] = W1[idx];
    }
    __syncthreads();
    const int wave = tid >> 5;            // 8 waves: 2 row-tiles x 4 col-tiles
    const int lane = tid & 31;
    const int wr = (wave >> 2) * 16;
    const int wc = (wave & 3) * 16;
    const int m  = lane & 15;
    const int kk = (lane >> 4) * 2;       // f32 16x16x4: lanes 0-15 hold K={0,1}, 16-31 K={2,3}
    v8f acc = {};
    for (int k0 = 0; k0 < IN_F; k0 += 4) {
        v2f a, b;
        a.x = xs[wr + m][k0 + kk];
        a.y = xs[wr + m][k0 + kk + 1];
        b.x = wt[k0 + kk][wc + m];
        b.y = wt[k0 + kk + 1][wc + m];
        acc = __builtin_amdgcn_wmma_f32_16x16x4_f32(false, a, false, b, (short)0, acc,
                                                    false, false);
    }
    const int rsub = (lane >> 4) * 8;     // D: VGPR r -> M = r (+8 for upper half-wave)
#pragma unroll
    for (int r = 0; r < 8; r++) {
        int grow = rowBase + wr + rsub + r;
        if (grow < NNODES) h1[grow * 64 + wc + m] = acc[r];
    }
}

// ---------------- attention coefficients, layer 1 ----------------
__global__ void k_coef1(const float* __restrict__ h1, const float* __restrict__ al,
                        const float* __restrict__ ar, float* el, float* er) {
    int i = blockIdx.x * blockDim.x + threadIdx.x;
    if (i >= NNODES * HEADS) return;
    int node = i >> 2, h = i & 3;
    const float* hv = h1 + node * 64 + h * HIDF;
    float sl = 0.f, sr = 0.f;
    for (int f = 0; f < HIDF; f++) { float v = hv[f]; sl += v * al[h * HIDF + f]; sr += v * ar[h * HIDF + f]; }
    el[i] = sl; er[i] = sr;
}

// ---------------- edge passes, layer 1 ----------------
__global__ void k_emax1(const int* __restrict__ src, const int* __restrict__ dst,
                        const float* __restrict__ el, const float* __restrict__ er, unsigned* m1) {
    int i = blockIdx.x * blockDim.x + threadIdx.x;
    if (i >= NEDGES * HEADS) return;
    int e = i >> 2, h = i & 3;
    int s = src[e], d = dst[e];
    float v = leaky(el[s * HEADS + h] + er[d * HEADS + h]);
    atomicMax(&m1[d * HEADS + h], f2o(v));
}
__global__ void k_eden1(const int* __restrict__ src, const int* __restrict__ dst,
                        const float* __restrict__ el, const float* __restrict__ er,
                        const unsigned* __restrict__ m1, float* den1) {
    int i = blockIdx.x * blockDim.x + threadIdx.x;
    if (i >= NEDGES * HEADS) return;
    int e = i >> 2, h = i & 3;
    int s = src[e], d = dst[e];
    float v = leaky(el[s * HEADS + h] + er[d * HEADS + h]);
    atomicAdd(&den1[d * HEADS + h], expf(v - o2f(m1[d * HEADS + h])));
}
__global__ void k_escat1(const int* __restrict__ src, const int* __restrict__ dst,
                         const float* __restrict__ el, const float* __restrict__ er,
                         const unsigned* __restrict__ m1, const float* __restrict__ den1,
                         const float* __restrict__ h1, float* agg1) {
    int i = blockIdx.x * blockDim.x + threadIdx.x;   // thread per (edge, feature) -> coalesced
    if (i >= NEDGES * 64) return;
    int e = i >> 6, c = i & 63, h = c >> 4;
    int s = src[e], d = dst[e];
    float v = leaky(el[s * HEADS + h] + er[d * HEADS + h]);
    float alpha = expf(v - o2f(m1[d * HEADS + h])) / den1[d * HEADS + h];
    atomicAdd(&agg1[d * 64 + c], h1[s * 64 + c] * alpha);
}

// ---------------- bias + relu (layer-1 output -> layer-2 input, in place) ----------------
__global__ void k_bias1(float* agg1, const float* __restrict__ b1) {
    int i = blockIdx.x * blockDim.x + threadIdx.x;
    if (i >= NNODES * 64) return;
    float v = agg1[i] + b1[i & 63];
    agg1[i] = v > 0.f ? v : 0.f;
}

// ---------------- GEMM2: h2[N,32] = z[N,64] @ W2[64,32]  (fp32 WMMA) ----------------
__global__ __launch_bounds__(256) void k_gemm2(const float* __restrict__ z,
                                               const float* __restrict__ W2,
                                               float* __restrict__ h2) {
    __shared__ float zs[64][64 + 1];
    __shared__ float wt[64][32 + 1];
    const int rowBase = blockIdx.x * 64;
    const int tid = threadIdx.x;
    for (int idx = tid; idx < 64 * 64; idx += 256) {
        int r = idx >> 6, c = idx & 63;
        int gr = rowBase + r;
        zs[r][c] = (gr < NNODES) ? z[gr * 64 + c] : 0.f;
    }
    for (int idx = tid; idx < 64 * 32; idx += 256) {
        int r = idx >> 5, c = idx & 31;
        wt[r][c] = W2[idx];
    }
    __syncthreads();
    const int wave = tid >> 5;            // 8 waves: 4 row-tiles x 2 col-tiles
    const int lane = tid & 31;
    const int wr = (wave >> 1) * 16;
    const int wc = (wave & 1) * 16;
    const int m  = lane & 15;
    const int kk = (lane >> 4) * 2;
    v8f acc = {};
    for (int k0 = 0; k0 < 64; k0 += 4) {
        v2f a, b;
        a.x = zs[wr + m][k0 + kk];
        a.y = zs[wr + m][k0 + kk + 1];
        b.x = wt[k0 + kk][wc + m];
        b.y = wt[k0 + kk + 1][wc + m];
        acc = __builtin_amdgcn_wmma_f32_16x16x4_f32(false, a, false, b, (short)0, acc,
                                                    false, false);
    }
    const int rsub = (lane >> 4) * 8;
#pragma unroll
    for (int r = 0; r < 8; r++) {
        int grow = rowBase + wr + rsub + r;
        if (grow < NNODES) h2[grow * OUTF + wc + m] = acc[r];
    }
}

// ---------------- attention coefficients, layer 2 (1 head) ----------------
__global__ void k_coef2(const float* __restrict__ h2, const float* __restrict__ al,
                        const float* __restrict__ ar, float* el, float* er) {
    int n = blockIdx.x * blockDim.x + threadIdx.x;
    if (n >= NNODES) return;
    float sl = 0.f, sr = 0.f;
    for (int c = 0; c < OUTF; c++) { float v = h2[n * OUTF + c]; sl += v * al[c]; sr += v * ar[c]; }
    el[n] = sl; er[n] = sr;
}

// ---------------- edge passes, layer 2 ----------------
__global__ void k_emax2(const int* __restrict__ src, const int* __restrict__ dst,
                        const float* __restrict__ el, const float* __restrict__ er, unsigned* m2) {
    int e = blockIdx.x * blockDim.x + threadIdx.x;
    if (e >= NEDGES) return;
    atomicMax(&m2[dst[e]], f2o(leaky(el[src[e]] + er[dst[e]])));
}
__global__ void k_eden2(const int* __restrict__ src, const int* __restrict__ dst,
                        const float* __restrict__ el, const float* __restrict__ er,
                        const unsigned* __restrict__ m2, float* den2) {
    int e = blockIdx.x * blockDim.x + threadIdx.x;
    if (e >= NEDGES) return;
    int d = dst[e];
    float v = leaky(el[src[e]] + er[d]);
    atomicAdd(&den2[d], expf(v - o2f(m2[d])));
}
__global__ void k_escat2(const int* __restrict__ src, const int* __restrict__ dst,
                         const float* __restrict__ el, const float* __restrict__ er,
                         const unsigned* __restrict__ m2, const float* __restrict__ den2,
                         const float* __restrict__ h2, float* agg2) {
    int i = blockIdx.x * blockDim.x + threadIdx.x;
    if (i >= NEDGES * OUTF) return;
    int e = i >> 5, c = i & 31;
    int s = src[e], d = dst[e];
    float v = leaky(el[s] + er[d]);
    float alpha = expf(v - o2f(m2[d])) / den2[d];
    atomicAdd(&agg2[d * OUTF + c], h2[s * OUTF + c] * alpha);
}

// ---------------- final mean over nodes ----------------
__global__ void k_reduce(const float* __restrict__ agg2, float* accum) {
    __shared__ float sm[256];
    int tid = threadIdx.x;
    float s = 0.f;
    // stride is a multiple of 32 so each thread stays on column (tid & 31)
    for (int t = blockIdx.x * 256 + tid; t < NNODES * OUTF; t += gridDim.x * 256) s += agg2[t];
    sm[tid] = s;
    __syncthreads();
    for (int off = 128; off >= 32; off >>= 1) {
        if (tid < off) sm[tid] += sm[tid + off];
        __syncthreads();
    }
    if (tid < 32) atomicAdd(&accum[tid], sm[tid]);
}
__global__ void k_final(const float* __restrict__ accum, const float* __restrict__ b2, float* out) {
    int c = threadIdx.x;
    if (c < OUTF) out[c] = accum[c] / (float)NNODES + b2[c];  // node-uniform bias folds into mean
}

extern "C" void kernel_launch(void* const* d_in, const int* in_sizes, int n_in,
                              void* d_out, int out_size, void* d_ws, size_t ws_size,
                              hipStream_t stream) {
    const float* x   = (const float*)d_in[0];
    const int*   src = (const int*)d_in[1];
    const int*   dst = (const int*)d_in[2];
    const float* W1  = (const float*)d_in[3];
    const float* al1 = (const float*)d_in[4];
    const float* ar1 = (const float*)d_in[5];
    const float* b1  = (const float*)d_in[6];
    const float* W2  = (const float*)d_in[7];
    const float* al2 = (const float*)d_in[8];
    const float* ar2 = (const float*)d_in[9];
    const float* b2  = (const float*)d_in[10];

    float* ws = (float*)d_ws;
    // layout (floats): h1[N*64] | el1[N*4] | er1[N*4] | m1[N*4] | den1[N*4] | agg1[N*64] | accum[32]
    float*    h1   = ws;
    float*    el1  = ws + NNODES * 64;
    float*    er1  = el1 + NNODES * HEADS;
    unsigned* m1   = (unsigned*)(er1 + NNODES * HEADS);
    float*    den1 = (float*)(m1 + NNODES * HEADS);
    float*    agg1 = den1 + NNODES * HEADS;       // becomes z (layer-2 input) after bias+relu
    float*    accum = agg1 + NNODES * 64;
    // layer-2 reuse: h1 region -> h2[N*32] + agg2[N*32]; scalar regions reused
    float*    h2   = ws;
    float*    agg2 = ws + NNODES * OUTF;
    float*    el2  = el1;
    float*    er2  = er1;
    unsigned* m2   = m1;
    float*    den2 = den1;

    k_init1<<<2048, 256, 0, stream>>>(agg1, m1, den1, accum);
    k_gemm1<<<(NNODES + 31) / 32, 256, 0, stream>>>(x, W1, h1);
    k_coef1<<<(NNODES * HEADS + 255) / 256, 256, 0, stream>>>(h1, al1, ar1, el1, er1);
    k_emax1<<<(NEDGES * HEADS + 255) / 256, 256, 0, stream>>>(src, dst, el1, er1, m1);
    k_eden1<<<(NEDGES * HEADS + 255) / 256, 256, 0, stream>>>(src, dst, el1, er1, m1, den1);
    k_escat1<<<(NEDGES * 64 + 255) / 256, 256, 0, stream>>>(src, dst, el1, er1, m1, den1, h1, agg1);
    k_bias1<<<(NNODES * 64 + 255) / 256, 256, 0, stream>>>(agg1, b1);
    k_init2<<<2048, 256, 0, stream>>>(agg2, m2, den2);
    k_gemm2<<<(NNODES + 63) / 64, 256, 0, stream>>>(agg1, W2, h2);
    k_coef2<<<(NNODES + 255) / 256, 256, 0, stream>>>(h2, al2, ar2, el2, er2);
    k_emax2<<<(NEDGES + 255) / 256, 256, 0, stream>>>(src, dst, el2, er2, m2);
    k_eden2<<<(NEDGES + 255) / 256, 256, 0, stream>>>(src, dst, el2, er2, m2, den2);
    k_escat2<<<(NEDGES * OUTF + 255) / 256, 256, 0, stream>>>(src, dst, el2, er2, m2, den2, h2, agg2);
    k_reduce<<<512, 256, 0, stream>>>(agg2, accum);
    k_final<<<1, 32, 0, stream>>>(accum, b2, (float*)d_out);
}